// KernelGenerator_65687229825204
// MI455X (gfx1250) — compile-verified
//
#include <hip/hip_runtime.h>
#include <math.h>

#define Bn   4
#define Cc   64
#define Hh   128
#define Ww   128
#define HW   (Hh*Ww)
#define NQ   35
#define MIDc 32
#define BCHW ((size_t)Bn*Cc*HW)

typedef float v2f __attribute__((ext_vector_type(2)));
typedef float v8f __attribute__((ext_vector_type(8)));

// ---------------- global mean over H,W per (b,c) ----------------
__global__ void reduce_mean_kernel(const float* __restrict__ x, float* __restrict__ xm) {
    __shared__ float sm[256];
    int bc = blockIdx.x;
    const float* p = x + (size_t)bc * HW;
    float s = 0.f;
    for (int i = threadIdx.x; i < HW; i += 256) s += p[i];
    sm[threadIdx.x] = s;
    __syncthreads();
    for (int off = 128; off > 0; off >>= 1) {
        if ((int)threadIdx.x < off) sm[threadIdx.x] += sm[threadIdx.x + off];
        __syncthreads();
    }
    if (threadIdx.x == 0) xm[bc] = sm[0] * (1.0f / HW);
}

// ---------------- spectral branch: fc1 + leaky + fc2 -> T slice ----------------
__global__ void spectral_kernel(const float* __restrict__ xm,
                                const float* __restrict__ fc1w, const float* __restrict__ fc1b,
                                const float* __restrict__ fc2w, const float* __restrict__ fc2b,
                                float* __restrict__ T, int kk, int qoff) {
    __shared__ float sxm[Cc];
    __shared__ float sg[MIDc];
    int b = blockIdx.x;
    if (threadIdx.x < Cc) sxm[threadIdx.x] = xm[b * Cc + threadIdx.x];
    __syncthreads();
    if (threadIdx.x < MIDc) {
        float acc = fc1b[threadIdx.x];
        for (int c = 0; c < Cc; c++) acc += fc1w[threadIdx.x * Cc + c] * sxm[c];
        sg[threadIdx.x] = acc > 0.f ? acc : 0.2f * acc;
    }
    __syncthreads();
    for (int r = threadIdx.x; r < Cc * kk; r += blockDim.x) {
        float acc = fc2b[r];
        for (int m = 0; m < MIDc; m++) acc += fc2w[r * MIDc + m] * sg[m];
        int c = r / kk, q = r % kk;
        T[(size_t)b * (Cc * NQ) + c * NQ + qoff + q] = acc;
    }
}

// ---------------- normalize T over last dim (35, ddof=1), in place ----------------
__global__ void normalize_T_kernel(float* __restrict__ T) {
    int idx = threadIdx.x;           // 256 threads = (b,c) rows
    float* row = T + (size_t)idx * NQ;
    float mean = 0.f;
    float v[NQ];
#pragma unroll
    for (int q = 0; q < NQ; q++) { v[q] = row[q]; mean += v[q]; }
    mean *= (1.0f / NQ);
    float var = 0.f;
#pragma unroll
    for (int q = 0; q < NQ; q++) { float d = v[q] - mean; var += d * d; }
    var *= (1.0f / (NQ - 1));
    float inv = 1.0f / (sqrtf(var) + 1e-10f);
#pragma unroll
    for (int q = 0; q < NQ; q++) row[q] = (v[q] - mean) * inv;
}

// ---------------- generic depthwise conv ----------------
__global__ void dw_conv_kernel(const float* __restrict__ in, const float* __restrict__ w,
                               const float* __restrict__ bias, float* __restrict__ out,
                               int CH, int k, int pad) {
    int gid = blockIdx.x * blockDim.x + threadIdx.x;
    int total = Bn * CH * HW;
    if (gid >= total) return;
    int p  = gid % HW;
    int ch = (gid / HW) % CH;
    int h = p / Ww, wc = p % Ww;
    const float* ip = in + (size_t)(gid - p);   // start of this (b,ch) plane
    const float* wp = w + ch * k * k;
    float acc = bias[ch];
    for (int dy = 0; dy < k; dy++) {
        int hy = h + dy - pad;
        if ((unsigned)hy >= (unsigned)Hh) continue;
        for (int dx = 0; dx < k; dx++) {
            int wx = wc + dx - pad;
            if ((unsigned)wx >= (unsigned)Ww) continue;
            acc += ip[hy * Ww + wx] * wp[dy * k + dx];
        }
    }
    out[gid] = acc;
}

// ---------------- pointwise (1x1) conv as GEMM via V_WMMA_F32_16X16X4_F32 ----------------
// out[((b*outCper)+coff+m)*HW + n] = act( sum_k w[m*Cin+k]*in[((b*Cin)+k)*HW+n] + bias[m] )
// block = 128 threads = 4 waves, each wave owns a 16-pixel N tile; grid=(HW/64, ceil(Cout/16), B)
__global__ void pw_conv_wmma_kernel(const float* __restrict__ in, const float* __restrict__ w,
                                    const float* __restrict__ bias, float* __restrict__ out,
                                    int Cin, int Cout, int outCper, int coff, int leaky) {
    __shared__ float Wlds[16 * 64];          // padded weight tile (M=16 x K<=64)
    int CinP = (Cin + 3) & ~3;
    int mt = blockIdx.y;
    for (int idx = threadIdx.x; idx < 16 * CinP; idx += blockDim.x) {
        int m = idx / CinP, kc = idx % CinP;
        int mg = mt * 16 + m;
        Wlds[m * CinP + kc] = (mg < Cout && kc < Cin) ? w[(size_t)mg * Cin + kc] : 0.f;
    }
    __syncthreads();

    int lane = threadIdx.x & 31;
    int wave = threadIdx.x >> 5;
    int half = lane >> 4;                    // 0: lanes 0-15, 1: lanes 16-31
    int mrow = lane & 15;
    int b = blockIdx.z;
    int n = (blockIdx.x * 4 + wave) * 16 + mrow;   // this lane's pixel column
    const float* inb = in + (size_t)b * Cin * HW;

    v8f acc = {};
    for (int kb = 0; kb < CinP; kb += 4) {
        int k0 = kb + 2 * half;              // A-matrix f32 16x4 lane layout
        int k1 = k0 + 1;
        v2f a, bb;
        a.x = Wlds[mrow * CinP + k0];
        a.y = Wlds[mrow * CinP + k1];
        bb.x = (k0 < Cin) ? inb[(size_t)k0 * HW + n] : 0.f;
        bb.y = (k1 < Cin) ? inb[(size_t)k1 * HW + n] : 0.f;
        acc = __builtin_amdgcn_wmma_f32_16x16x4_f32(
            /*neg_a=*/false, a, /*neg_b=*/false, bb,
            /*c_mod=*/(short)0, acc, /*reuse_a=*/false, /*reuse_b=*/false);
    }

    float* ob = out + ((size_t)b * outCper + coff) * HW + n;
#pragma unroll
    for (int r = 0; r < 8; r++) {
        int m = r + (half ? 8 : 0);          // D layout: VGPR r -> M=r / M=r+8
        int mg = mt * 16 + m;
        if (mg < Cout) {
            float v = acc[r] + bias[mg];
            if (leaky && v < 0.f) v *= 0.2f;
            ob[(size_t)mg * HW] = v;
        }
    }
}

// ---------------- fused: normalize S over 35 ch + patch einsum, all 3 outputs ----------------
__global__ void final_einsum_kernel(const float* __restrict__ x, const float* __restrict__ S,
                                    const float* __restrict__ T, float* __restrict__ out) {
    __shared__ float sT[Cc * NQ];            // 8960 B
    __shared__ float sx[20 * 20];            // x tile + halo 2
    int b = blockIdx.z;
    int tx = threadIdx.x & 15, ty = threadIdx.x >> 4;
    int w0 = blockIdx.x * 16, h0 = blockIdx.y * 16;
    int h = h0 + ty, wc = w0 + tx;

    for (int i = threadIdx.x; i < Cc * NQ; i += 256) sT[i] = T[(size_t)b * Cc * NQ + i];

    // per-pixel S normalization (ddof=1) kept in registers
    float sk[NQ];
    const float* Sp = S + (size_t)b * NQ * HW + h * Ww + wc;
    float mean = 0.f;
#pragma unroll
    for (int q = 0; q < NQ; q++) { sk[q] = Sp[(size_t)q * HW]; mean += sk[q]; }
    mean *= (1.0f / NQ);
    float var = 0.f;
#pragma unroll
    for (int q = 0; q < NQ; q++) { float d = sk[q] - mean; var += d * d; }
    var *= (1.0f / (NQ - 1));
    float inv = 1.0f / (sqrtf(var) + 1e-10f);
#pragma unroll
    for (int q = 0; q < NQ; q++) sk[q] = (sk[q] - mean) * inv;

    for (int c = 0; c < Cc; c++) {
        __syncthreads();
        const float* xp = x + ((size_t)b * Cc + c) * HW;
        for (int i = threadIdx.x; i < 20 * 20; i += 256) {
            int r = i / 20, cc = i % 20;
            int gh = h0 + r - 2, gw = w0 + cc - 2;
            sx[i] = ((unsigned)gh < (unsigned)Hh && (unsigned)gw < (unsigned)Ww)
                        ? xp[gh * Ww + gw] : 0.f;
        }
        __syncthreads();
        const float* tc = &sT[c * NQ];
        int cy = ty + 2, cx = tx + 2;
        float a0 = sx[cy * 20 + cx] * sk[0] * tc[0];
        float a1 = 0.f, a2 = 0.f;
#pragma unroll
        for (int dy = 0; dy < 3; dy++)
#pragma unroll
            for (int dx = 0; dx < 3; dx++) {
                int q = 1 + dy * 3 + dx;
                a1 += sx[(cy + dy - 1) * 20 + (cx + dx - 1)] * sk[q] * tc[q];
            }
#pragma unroll
        for (int dy = 0; dy < 5; dy++)
#pragma unroll
            for (int dx = 0; dx < 5; dx++) {
                int q = 10 + dy * 5 + dx;
                a2 += sx[(cy + dy - 2) * 20 + (cx + dx - 2)] * sk[q] * tc[q];
            }
        size_t oo = ((size_t)b * Cc + c) * HW + h * Ww + wc;
        out[oo]            = a0;
        out[BCHW + oo]     = a1;
        out[2 * BCHW + oo] = a2;
    }
}

extern "C" void kernel_launch(void* const* d_in, const int* in_sizes, int n_in,
                              void* d_out, int out_size, void* d_ws, size_t ws_size,
                              hipStream_t stream) {
    (void)in_sizes; (void)n_in; (void)out_size; (void)ws_size;
    const float* x = (const float*)d_in[0];
    const float* y = (const float*)d_in[1];
    float* ws = (float*)d_ws;

    // workspace layout (floats)
    float* xm = ws;                                   // 256
    float* T  = xm + 256;                             // B*C*35 = 8960
    float* S  = T + (size_t)Bn * Cc * NQ;             // B*35*HW = 2,293,760
    float* t0 = S + (size_t)Bn * NQ * HW;             // B*C*HW  = 4,194,304
    float* t1 = t0 + BCHW;                            // B*25*HW = 1,638,400
    float* t2 = t1 + (size_t)Bn * 25 * HW;            // B*25*HW
    float* out = (float*)d_out;

    reduce_mean_kernel<<<Bn * Cc, 256, 0, stream>>>(x, xm);

    const int KS[3]   = {1, 3, 5};
    const int PADS[3] = {0, 1, 2};
    const int QOFF[3] = {0, 1, 10};

    // spectral branch: inputs 38 + i*4 + {fc1_w, fc1_b, fc2_w, fc2_b}
    for (int i = 0; i < 3; i++) {
        int kk = KS[i] * KS[i];
        spectral_kernel<<<Bn, 256, 0, stream>>>(
            xm,
            (const float*)d_in[38 + i * 4 + 0], (const float*)d_in[38 + i * 4 + 1],
            (const float*)d_in[38 + i * 4 + 2], (const float*)d_in[38 + i * 4 + 3],
            T, kk, QOFF[i]);
    }
    normalize_T_kernel<<<1, 256, 0, stream>>>(T);

    // spatial branches: inputs 2 + i*12 + {dw1_w,dw1_b,pw1_w,pw1_b,dw2_w,dw2_b,pw2_w,pw2_b,dw3_w,dw3_b,pw3_w,pw3_b}
    for (int i = 0; i < 3; i++) {
        int k = KS[i], pad = PADS[i], kk = k * k;
        const float** P = (const float**)&d_in[2 + i * 12];
        const float *dw1w = P[0], *dw1b = P[1], *pw1w = P[2], *pw1b = P[3];
        const float *dw2w = P[4], *dw2b = P[5], *pw2w = P[6], *pw2b = P[7];
        const float *dw3w = P[8], *dw3b = P[9], *pw3w = P[10], *pw3b = P[11];

        int totC  = Bn * Cc * HW;
        int totKK = Bn * kk * HW;
        dim3 gpw(HW / 64, (kk + 15) / 16, Bn);

        dw_conv_kernel<<<(totC + 255) / 256, 256, 0, stream>>>(y, dw1w, dw1b, t0, Cc, k, pad);
        pw_conv_wmma_kernel<<<gpw, 128, 0, stream>>>(t0, pw1w, pw1b, t1, Cc, kk, kk, 0, 0);
        dw_conv_kernel<<<(totKK + 255) / 256, 256, 0, stream>>>(t1, dw2w, dw2b, t2, kk, k, pad);
        pw_conv_wmma_kernel<<<gpw, 128, 0, stream>>>(t2, pw2w, pw2b, t1, kk, kk, kk, 0, 1);
        dw_conv_kernel<<<(totKK + 255) / 256, 256, 0, stream>>>(t1, dw3w, dw3b, t2, kk, k, pad);
        pw_conv_wmma_kernel<<<gpw, 128, 0, stream>>>(t2, pw3w, pw3b, S, kk, kk, NQ, QOFF[i], 0);
    }

    dim3 gf(Ww / 16, Hh / 16, Bn);
    final_einsum_kernel<<<gf, 256, 0, stream>>>(x, S, T, out);
}